// ExampleModel_1992864825952
// MI455X (gfx1250) — compile-verified
//
#include <hip/hip_runtime.h>
#include <math.h>

// Problem constants (match reference)
#define BB 8
#define SS 1024
#define DD 2048
#define NTOK (BB * SS)          // 8192 tokens
#define LDS_STRIDE (DD + 4)     // pad LDS B-columns to dodge bank conflicts

typedef __attribute__((ext_vector_type(2))) float v2f;
typedef __attribute__((ext_vector_type(8))) float v8f;

// ---------------------------------------------------------------------------
// Kernel 1: wsum[e,d] = sum_h W[e,d,h]  (rows 0..2D-1),  bsum[e] = sum_h b[e,h]
// One wave per 2048-length row; coalesced float4 loads; shfl tree reduce.
// ---------------------------------------------------------------------------
__global__ void __launch_bounds__(256) reduce_w_kernel(
    const float* __restrict__ W, const float* __restrict__ b,
    float* __restrict__ c4T, float* __restrict__ bsum) {
  const int row  = (blockIdx.x * blockDim.x + threadIdx.x) >> 5;
  const int lane = threadIdx.x & 31;
  const int nW = 2 * DD;          // 4096 W rows
  if (row >= nW + 2) return;

  const float* src = (row < nW) ? (W + (size_t)row * DD)
                                : (b + (size_t)(row - nW) * DD);
  float s = 0.f;
  const float4* src4 = (const float4*)src;   // 512 float4 per row
#pragma unroll
  for (int i = 0; i < 16; ++i) {
    float4 v = src4[lane + i * 32];
    s += (v.x + v.y) + (v.z + v.w);
  }
#pragma unroll
  for (int off = 16; off > 0; off >>= 1)
    s += __shfl_down(s, off, 32);

  if (lane == 0) {
    if (row < nW) c4T[row] = s;              // c4T rows 0,1 = wsum[0], wsum[1]
    else          bsum[row - nW] = s;
  }
}

// ---------------------------------------------------------------------------
// Kernel 2: pack gate weights as B-matrix columns 2,3:  c4T[(2+e)*D + d] = wg[d,e]
// ---------------------------------------------------------------------------
__global__ void pack_wg_kernel(const float* __restrict__ wg,
                               float* __restrict__ c4T) {
  int d = blockIdx.x * blockDim.x + threadIdx.x;
  if (d < DD) {
    c4T[2 * DD + d] = wg[d * 2 + 0];
    c4T[3 * DD + d] = wg[d * 2 + 1];
  }
}

// ---------------------------------------------------------------------------
// Kernel 3: WMMA token kernel. Each wave owns a 16-token tile and computes
// C[16x16] = A[16xD] * B[Dx16] via V_WMMA_F32_16X16X4_F32 (K-steps of 4,
// software-pipelined 32-K chunks with ping-pong register double buffering).
// Only B columns 0..3 are live (wsum0, wsum1, wg0, wg1); columns are
// replicated with n&3 so every lane feeds valid data (EXEC all-ones) and
// the extra columns are simply ignored.
// ---------------------------------------------------------------------------
__global__ void __launch_bounds__(64) moe_token_kernel(
    const float* __restrict__ x, const float* __restrict__ c4T,
    const float* __restrict__ bsum, float* __restrict__ y) {
  __shared__ float sB[4 * LDS_STRIDE];       // padded B columns, ~32.8 KB
  __shared__ float stage[2][16][4];          // per-wave C unload

  const int tid = threadIdx.x;

  // Stage the 4 packed weight columns into LDS (float4, padded stride).
#pragma unroll
  for (int i = 0; i < 32; ++i) {
    int g4  = tid + i * 64;                  // float4 index over 4*D/4 = 2048
    int col = g4 >> 9;                       // / 512
    int d4  = g4 & 511;
    float4 v = ((const float4*)c4T)[g4];
    *(float4*)&sB[col * LDS_STRIDE + d4 * 4] = v;
  }
  __syncthreads();

  const int wave = tid >> 5;
  const int lane = tid & 31;
  const int m    = lane & 15;                // A-row / token within tile
  const int kp   = (lane >> 4) << 1;         // K sub-pair: 0 or 2
  const int tile = blockIdx.x * 2 + wave;    // 512 tiles total

  const float* xrow = x + (size_t)(tile * 16 + m) * DD;
  const float* sBn  = sB + (size_t)(lane & 3) * LDS_STRIDE;  // replicate cols

  v8f c = {};

  auto load8 = [&](v2f* av, v2f* bv, int k0) {
#pragma unroll
    for (int j = 0; j < 8; ++j) {
      av[j] = *(const v2f*)(xrow + k0 + j * 4 + kp);
      bv[j] = *(const v2f*)(sBn  + k0 + j * 4 + kp);
    }
  };
  auto compute8 = [&](const v2f* av, const v2f* bv) {
#pragma unroll
    for (int j = 0; j < 8; ++j)
      c = __builtin_amdgcn_wmma_f32_16x16x4_f32(
          /*neg_a=*/false, av[j], /*neg_b=*/false, bv[j],
          /*c_mod=*/(short)0, c, /*reuse_a=*/false, /*reuse_b=*/false);
  };

  // Ping-pong pipeline: loads for chunk n+1 issue before WMMAs of chunk n.
  v2f a0[8], b0[8], a1[8], b1[8];
  load8(a0, b0, 0);
  int k0 = 0;
  for (; k0 < DD - 64; k0 += 64) {
    load8(a1, b1, k0 + 32);
    compute8(a0, b0);
    load8(a0, b0, k0 + 64);
    compute8(a1, b1);
  }
  load8(a1, b1, k0 + 32);                    // final chunk @ 2016
  compute8(a0, b0);                          // chunk @ 1984
  compute8(a1, b1);                          // chunk @ 2016

  // Unload the 4 live C columns: lane holds (M = j + 8*(lane>=16), N = lane%16)
  const int nn = lane & 15;
  if (nn < 4) {
    const int mbase = (lane >> 4) * 8;
#pragma unroll
    for (int j = 0; j < 8; ++j)
      stage[wave][mbase + j][nn] = c[j];
  }
  __syncthreads();

  if (lane < 16) {
    float xw0 = stage[wave][lane][0];
    float xw1 = stage[wave][lane][1];
    float l0  = stage[wave][lane][2];
    float l1  = stage[wave][lane][3];
    int   idx  = (l1 > l0) ? 1 : 0;                       // argmax (first on tie)
    float gate = 1.f / (1.f + expf(-fabsf(l1 - l0)));     // top-1 softmax prob
    float xw   = idx ? xw1 : xw0;
    y[tile * 16 + lane] = gate * (xw + bsum[idx]);
  }
}

// ---------------------------------------------------------------------------
// Kernel 4: log_softmax over S per batch row.  8 rows x 1024.
// ---------------------------------------------------------------------------
__global__ void __launch_bounds__(256) logsoftmax_kernel(
    const float* __restrict__ yin, float* __restrict__ out) {
  __shared__ float red[256];
  const int t = threadIdx.x;
  const float* row = yin + (size_t)blockIdx.x * SS;
  float v0 = row[t], v1 = row[t + 256], v2 = row[t + 512], v3 = row[t + 768];

  float m = fmaxf(fmaxf(v0, v1), fmaxf(v2, v3));
  red[t] = m;
  __syncthreads();
#pragma unroll
  for (int off = 128; off > 0; off >>= 1) {
    if (t < off) red[t] = fmaxf(red[t], red[t + off]);
    __syncthreads();
  }
  float rowmax = red[0];
  __syncthreads();

  float s = expf(v0 - rowmax) + expf(v1 - rowmax) +
            expf(v2 - rowmax) + expf(v3 - rowmax);
  red[t] = s;
  __syncthreads();
#pragma unroll
  for (int off = 128; off > 0; off >>= 1) {
    if (t < off) red[t] += red[t + off];
    __syncthreads();
  }
  float lse = rowmax + logf(red[0]);

  float* orow = out + (size_t)blockIdx.x * SS;
  orow[t]       = v0 - lse;
  orow[t + 256] = v1 - lse;
  orow[t + 512] = v2 - lse;
  orow[t + 768] = v3 - lse;
}

// ---------------------------------------------------------------------------
extern "C" void kernel_launch(void* const* d_in, const int* in_sizes, int n_in,
                              void* d_out, int out_size, void* d_ws, size_t ws_size,
                              hipStream_t stream) {
  (void)in_sizes; (void)n_in; (void)out_size; (void)ws_size;
  const float* x  = (const float*)d_in[0];   // [B,S,D]
  const float* wg = (const float*)d_in[1];   // [D,E]
  const float* W  = (const float*)d_in[2];   // [E,D,D]
  const float* b  = (const float*)d_in[3];   // [E,D]
  float* out = (float*)d_out;                // [B,S]
  float* ws  = (float*)d_ws;

  float* c4T  = ws;                          // 4*D floats (B-matrix columns)
  float* bsum = ws + 4 * DD;                 // 2 floats
  float* yb   = ws + 4 * DD + 8;             // NTOK floats (16B aligned)

  // rows = 2*D (wsum) + 2 (bsum); 8 waves per 256-thread block
  reduce_w_kernel<<<(2 * DD + 2 + 7) / 8, 256, 0, stream>>>(W, b, c4T, bsum);
  pack_wg_kernel<<<(DD + 255) / 256, 256, 0, stream>>>(wg, c4T);
  moe_token_kernel<<<NTOK / 32, 64, 0, stream>>>(x, c4T, bsum, yb);
  logsoftmax_kernel<<<BB, 256, 0, stream>>>(yb, out);
}